// CombinedLoss_86792699118042
// MI455X (gfx1250) — compile-verified
//
#include <hip/hip_runtime.h>
#include <hip/hip_bf16.h>
#include <math.h>

typedef __attribute__((ext_vector_type(2))) float v2f;
typedef __attribute__((ext_vector_type(8))) float v8f;

// Butterfly-min within each 16-lane half, add per-row |x|^2 (pulled out of the
// min), and store 16 row results from lanes 0 / 16.
__device__ __forceinline__ void reduce_add_store(v8f rmin, float xsq, bool hi,
                                                 int lane, float* __restrict__ dst_base) {
#pragma unroll
    for (int i = 0; i < 8; ++i) {
        float v = rmin[i];
        v = fminf(v, __shfl_xor(v, 8, 32));
        v = fminf(v, __shfl_xor(v, 4, 32));
        v = fminf(v, __shfl_xor(v, 2, 32));
        v = fminf(v, __shfl_xor(v, 1, 32));
        // |x|^2 of row (i + (hi?8:0)) lives in lane i (lo) / lane i+24 (hi dup)
        float xr = __shfl(xsq, hi ? (i + 24) : i, 32);
        rmin[i] = v + xr;
    }
    if (lane == 0 || lane == 16) {
        float* dst = dst_base + (hi ? 8 : 0);
#pragma unroll
        for (int i = 0; i < 8; ++i) dst[i] = rmin[i];
    }
}

// One wave owns TWO 16-row strips (32 rows) of X for one batch and scans all
// M columns of Y in 16-wide tiles. Per tile, one B-operand build feeds two
// WMMAs:
//   D = A(16x4) * B(4x16) + 0        (inline-0 accumulator, no C-init movs)
//   A = [-2x0,-2x1,-2x2, 1],  B = [y0,y1,y2,|y|^2]   ==>  D = |y|^2 - 2 x.y
// Row mins of D accumulate in registers; |x|^2 is added after the loop.
__global__ void __launch_bounds__(256)
chamfer_rowmin_kernel(const float* __restrict__ X, const float* __restrict__ Y,
                      float* __restrict__ rowmin_out, int B, int N, int M) {
    const int  lane = threadIdx.x & 31;
    const bool hi   = lane >= 16;           // lanes 16..31 hold K=2 / K=3 slots
    const int  sub  = lane & 15;
    const int  wave = blockIdx.x * (blockDim.x >> 5) + (threadIdx.x >> 5);
    const int  ntiles = N >> 5;             // 32 rows per wave
    const int  b   = wave / ntiles;
    const int  nt  = wave - b * ntiles;
    if (b >= B) return;                     // wave-uniform: EXEC stays all-ones
    const int  n0  = nt << 5;

    // ---- Build two A operands (16x4 f32 each, 2 VGPRs) once ----
    const float* xp0 = X + ((size_t)b * N + (n0 + sub)) * 3;
    const float* xp1 = xp0 + 48;            // next 16 rows
    const float x00 = xp0[0], x01 = xp0[1], x02 = xp0[2];
    const float x10 = xp1[0], x11 = xp1[1], x12 = xp1[2];
    const float xsq0 = x00 * x00 + x01 * x01 + x02 * x02;
    const float xsq1 = x10 * x10 + x11 * x11 + x12 * x12;
    v2f a0, a1;
    a0.x = hi ? (-2.0f * x02) : (-2.0f * x00);   // K=2 : K=0
    a0.y = hi ? 1.0f          : (-2.0f * x01);   // K=3 : K=1
    a1.x = hi ? (-2.0f * x12) : (-2.0f * x10);
    a1.y = hi ? 1.0f          : (-2.0f * x11);

    v8f rmin0, rmin1;
#pragma unroll
    for (int i = 0; i < 8; ++i) { rmin0[i] = 3.4e38f; rmin1[i] = 3.4e38f; }

    const v8f zeroC = {};                   // lowers to inline-0 SRC2
    const float* Yb = Y + (size_t)b * (size_t)M * 3;
    for (int m0 = 0; m0 < M; m0 += 16) {
        // Lanes l and l+16 load the same point; no cross-lane shuffles needed.
        const float* yp = Yb + (size_t)(m0 + sub) * 3;
        const float y0 = yp[0], y1 = yp[1], y2 = yp[2];
        const float ysq = y0 * y0 + y1 * y1 + y2 * y2;
        v2f bb;
        bb.x = hi ? y2  : y0;               // K=2 : K=0
        bb.y = hi ? ysq : y1;               // K=3 : K=1
        v8f c0 = __builtin_amdgcn_wmma_f32_16x16x4_f32(
                     false, a0, false, bb, (short)0, zeroC, false, false);
        v8f c1 = __builtin_amdgcn_wmma_f32_16x16x4_f32(
                     false, a1, false, bb, (short)0, zeroC, false, false);
#pragma unroll
        for (int i = 0; i < 8; ++i) {
            rmin0[i] = fminf(rmin0[i], c0[i]);
            rmin1[i] = fminf(rmin1[i], c1[i]);
        }
    }

    float* base = rowmin_out + (size_t)b * N + n0;
    reduce_add_store(rmin0, xsq0, hi, lane, base);
    reduce_add_store(rmin1, xsq1, hi, lane, base + 16);
}

// Single block: deterministic tree reduction of sqrt(max(d^2,0)) over both
// directions + tiny registration loss; writes (L_R+L_CD, L_R, L_CD).
__global__ void __launch_bounds__(256)
finalize_kernel(const float* __restrict__ mins, int BN, int BM,
                float invBN, float invBM,
                const float* __restrict__ R,  const float* __restrict__ t,
                const float* __restrict__ S,  const float* __restrict__ Rg,
                const float* __restrict__ tg, const float* __restrict__ Sg,
                int B, float* __restrict__ out) {
    __shared__ float sdata[256];
    __shared__ float lr[32];
    const int tid = threadIdx.x;

    float s1 = 0.0f, s2 = 0.0f;
    for (int i = tid; i < BN; i += 256) {
        float v = mins[i];
        s1 += sqrtf(v > 0.0f ? v : 0.0f);
    }
    for (int i = tid; i < BM; i += 256) {
        float v = mins[BN + i];
        s2 += sqrtf(v > 0.0f ? v : 0.0f);
    }
    sdata[tid] = s1 * invBN + s2 * invBM;
    __syncthreads();
#pragma unroll
    for (int stride = 128; stride > 0; stride >>= 1) {
        if (tid < stride) sdata[tid] += sdata[tid + stride];
        __syncthreads();
    }

    if (tid < B) {
        const float* Rb  = R  + tid * 9;
        const float* Rgb = Rg + tid * 9;
        float acc = 0.0f;
#pragma unroll
        for (int i = 0; i < 3; ++i)
#pragma unroll
            for (int k = 0; k < 3; ++k) {
                float d = 0.0f;
#pragma unroll
                for (int j = 0; j < 3; ++j) d += Rgb[j * 3 + i] * Rb[j * 3 + k];
                if (i == k) d -= 1.0f;
                acc += d * d;
            }
#pragma unroll
        for (int j = 0; j < 3; ++j) {
            float dt = tg[tid * 3 + j] - t[tid * 3 + j];
            acc += dt * dt;
        }
        float ds = Sg[tid] - S[tid];
        acc += ds * ds;
        lr[tid] = acc;
    }
    __syncthreads();

    if (tid == 0) {
        float L_R = 0.0f;
        for (int b = 0; b < B; ++b) L_R += lr[b];
        L_R /= (float)B;
        float L_CD = sdata[0];
        out[0] = L_R + L_CD;
        out[1] = L_R;
        out[2] = L_CD;
    }
}

extern "C" void kernel_launch(void* const* d_in, const int* in_sizes, int n_in,
                              void* d_out, int out_size, void* d_ws, size_t ws_size,
                              hipStream_t stream) {
    const float* R  = (const float*)d_in[0];
    const float* t  = (const float*)d_in[1];
    const float* S  = (const float*)d_in[2];
    const float* Rg = (const float*)d_in[3];
    const float* tg = (const float*)d_in[4];
    const float* Sg = (const float*)d_in[5];
    const float* TX = (const float*)d_in[6];
    const float* Y  = (const float*)d_in[7];
    float* out = (float*)d_out;

    const int B = in_sizes[2];                 // S is [B]
    const int N = in_sizes[6] / (3 * B);       // T_X is [B,N,3]
    const int M = in_sizes[7] / (3 * B);       // Y   is [B,M,3]

    float* ws_row = (float*)d_ws;              // B*N row-min d^2 (x -> y)
    float* ws_col = ws_row + (size_t)B * N;    // B*M col-min d^2 (y -> x)

    const int wavesXY = B * (N >> 5);          // 32 rows per wave
    const int wavesYX = B * (M >> 5);
    const int blkXY = (wavesXY + 7) / 8;       // 8 waves / 256-thread block
    const int blkYX = (wavesYX + 7) / 8;

    chamfer_rowmin_kernel<<<blkXY, 256, 0, stream>>>(TX, Y, ws_row, B, N, M);
    chamfer_rowmin_kernel<<<blkYX, 256, 0, stream>>>(Y, TX, ws_col, B, M, N);

    const float invBN = 1.0f / ((float)B * (float)N);
    const float invBM = 1.0f / ((float)B * (float)M);
    finalize_kernel<<<1, 256, 0, stream>>>(ws_row, B * N, B * M, invBN, invBM,
                                           R, t, S, Rg, tg, Sg, B, out);
}